// LiftSplatShoot_3375844294851
// MI455X (gfx1250) — compile-verified
//
#include <hip/hip_runtime.h>
#include <stdint.h>

// ---------------------------------------------------------------------------
// LSS voxel pooling for MI455X (gfx1250, wave32).
// Grid constants from reference: DX=(0.5,0.5,20), BX=(-49.75,-49.75,0),
// NX=(200,200,1).  bin = (b*200+ix)*200+iy,
// out[(b*64+c)*40000 + ix*200 + iy] = sum of x[p, c] over points p in bin.
//
// Pipeline: histogram -> scan -> counting-sort point lists -> per-bin gather
// with async global->LDS staging (ASYNCcnt) -> coalesced channel-major store.
// ---------------------------------------------------------------------------

#define LSS_CHUNK 4   // async-staged feature rows in flight per wave

__device__ __forceinline__ int lss_bin_of(float gx, float gy, float gz, int b) {
  // replicate reference exactly: trunc-toward-zero of (g - (BX - DX/2)) / DX
  int ix = (int)((gx - (-50.0f)) / 0.5f);
  int iy = (int)((gy - (-50.0f)) / 0.5f);
  int iz = (int)((gz - (-10.0f)) / 20.0f);
  bool kept = (ix >= 0) & (ix < 200) & (iy >= 0) & (iy < 200) & (iz == 0);
  return kept ? ((b * 200 + ix) * 200 + iy) : -1;
}

// ---------------- pass 4 (defined first for disasm visibility) --------------
// gather + async-LDS staged accumulate + coalesced write.
// One block per (b, xi) row: 8 waves, one wave per bin (yi = wave + 8*t), so
// bin/count/list values are wave-uniform; readfirstlane makes that explicit so
// loop control is scalar (s_cmp/s_cbranch) and EXEC stays all-ones around the
// async ops.  Feature rows (256B) are staged with global_load_async_to_lds_b64
// (32 lanes x 8B), consumed after s_wait_asynccnt; s_wait_dscnt guards the WAR
// on slot reuse.  Results transpose through a stride-66 (bank-conflict-free)
// LDS tile so the channel-major output writes are fully coalesced.
__global__ __launch_bounds__(256) void lss_gather(const float* __restrict__ x,
                                                  const int* __restrict__ ptlist,
                                                  const int* __restrict__ offs,
                                                  const int* __restrict__ counts,
                                                  float* __restrict__ out) {
  __shared__ __align__(16) float sums[200 * 66];          // 52.8 KB
  __shared__ __align__(16) float stage[8][LSS_CHUNK][64]; // 8 KB staging
  const int tid  = threadIdx.x;
  const int wv   = tid >> 5;
  const int lane = tid & 31;
  const int blk  = blockIdx.x;
  const int b    = blk / 200;
  const int xi   = blk - b * 200;
  const int binBase = b * 40000 + xi * 200;
  const unsigned stageBase = (unsigned)(uintptr_t)(&stage[wv][0][0]);
  const uint64_t xbase     = (uint64_t)(uintptr_t)x;
  const unsigned laneOff   = (unsigned)lane << 3;

  for (int t = 0; t < 25; ++t) {
    const int yi  = wv + (t << 3);      // covers yi = 0..199 across 8 waves
    const int bin = binBase + yi;
    // wave-uniform by construction; make it explicit for scalar control flow
    const int s = __builtin_amdgcn_readfirstlane(offs[bin]);
    const int k = __builtin_amdgcn_readfirstlane(counts[bin]);
    float a0 = 0.0f, a1 = 0.0f;
    for (int base = 0; base < k; base += LSS_CHUNK) {
      const int m = min(LSS_CHUNK, k - base);
      // previous chunk's LDS reads must land before the engine overwrites slots
      asm volatile("s_wait_dscnt 0x0" ::: "memory");
      for (int j = 0; j < m; ++j) {
        const int p = __builtin_amdgcn_readfirstlane(ptlist[s + base + j]);
        const unsigned goff = ((unsigned)p << 8) + laneOff;
        const unsigned loff = stageBase + ((unsigned)j << 8) + laneOff;
        asm volatile("global_load_async_to_lds_b64 %0, %1, %2"
                     :: "v"(loff), "v"(goff), "s"(xbase)
                     : "memory");
      }
      asm volatile("s_wait_asynccnt 0x0" ::: "memory");
      for (int j = 0; j < m; ++j) {
        const float2 v = *(const float2*)(&stage[wv][j][lane << 1]);
        a0 += v.x;
        a1 += v.y;
      }
    }
    *(float2*)(&sums[yi * 66 + (lane << 1)]) = make_float2(a0, a1);
  }
  __syncthreads();

  // Phase B: channel-major coalesced write-out
  const int obase = (b * 64) * 40000 + xi * 200;
  for (int i = tid; i < 200 * 64; i += 256) {
    const int c  = i / 200;
    const int yi = i - c * 200;
    out[obase + c * 40000 + yi] = sums[yi * 66 + c];
  }
}

// ---------------- pass 0: zero the histogram ----------------
__global__ __launch_bounds__(256) void lss_zero_i32(int* __restrict__ p, int n) {
  int i = blockIdx.x * 256 + threadIdx.x;
  if (i < n) p[i] = 0;
}

// ---------------- pass 1: bin points + histogram ----------------
__global__ __launch_bounds__(256) void lss_bin_count(const float* __restrict__ geom,
                                                     int* __restrict__ pbin,
                                                     int* __restrict__ counts,
                                                     int np, int perB) {
  int p = blockIdx.x * 256 + threadIdx.x;
  if (p >= np) return;
  float gx = geom[3 * p + 0];
  float gy = geom[3 * p + 1];
  float gz = geom[3 * p + 2];
  int b = p / perB;
  int bin = lss_bin_of(gx, gy, gz, b);
  pbin[p] = bin;
  if (bin >= 0) atomicAdd(&counts[bin], 1);
}

// ---------------- pass 2a: per-1024-block scan ----------------
__global__ __launch_bounds__(256) void lss_scan1(const int* __restrict__ in,
                                                 int* __restrict__ lexcl,
                                                 int* __restrict__ bsum, int n) {
  __shared__ int sh[256];
  const int tid  = threadIdx.x;
  const int base = blockIdx.x * 1024 + tid * 4;
  int v0 = (base + 0 < n) ? in[base + 0] : 0;
  int v1 = (base + 1 < n) ? in[base + 1] : 0;
  int v2 = (base + 2 < n) ? in[base + 2] : 0;
  int v3 = (base + 3 < n) ? in[base + 3] : 0;
  const int ts = v0 + v1 + v2 + v3;
  sh[tid] = ts;
  __syncthreads();
  for (int o = 1; o < 256; o <<= 1) {
    int t = (tid >= o) ? sh[tid - o] : 0;
    __syncthreads();
    sh[tid] += t;
    __syncthreads();
  }
  int e = sh[tid] - ts;  // exclusive prefix of this thread's 4-group
  if (base + 0 < n) lexcl[base + 0] = e;
  if (base + 1 < n) lexcl[base + 1] = e + v0;
  if (base + 2 < n) lexcl[base + 2] = e + v0 + v1;
  if (base + 3 < n) lexcl[base + 3] = e + v0 + v1 + v2;
  if (tid == 255) bsum[blockIdx.x] = sh[255];
}

// ---------------- pass 2b: scan the block sums (n <= 512) ----------------
__global__ __launch_bounds__(512) void lss_scan2(const int* __restrict__ bsum,
                                                 int* __restrict__ boff, int n) {
  __shared__ int sh[512];
  const int tid = threadIdx.x;
  int v = (tid < n) ? bsum[tid] : 0;
  sh[tid] = v;
  __syncthreads();
  for (int o = 1; o < 512; o <<= 1) {
    int t = (tid >= o) ? sh[tid - o] : 0;
    __syncthreads();
    sh[tid] += t;
    __syncthreads();
  }
  if (tid < n) boff[tid] = sh[tid] - v;
}

// ---------------- pass 2c: finalize offsets, seed cursors ----------------
__global__ __launch_bounds__(256) void lss_scan3(const int* __restrict__ lexcl,
                                                 const int* __restrict__ boff,
                                                 int* __restrict__ offs,
                                                 int* __restrict__ cursor, int n) {
  int i = blockIdx.x * 256 + threadIdx.x;
  if (i >= n) return;
  int o = lexcl[i] + boff[i >> 10];
  offs[i]   = o;
  cursor[i] = o;
}

// ---------------- pass 3: scatter point indices into bin lists ----------------
__global__ __launch_bounds__(256) void lss_scatter(const int* __restrict__ pbin,
                                                   int* __restrict__ cursor,
                                                   int* __restrict__ ptlist, int np) {
  int p = blockIdx.x * 256 + threadIdx.x;
  if (p >= np) return;
  int bin = pbin[p];
  if (bin < 0) return;
  int pos = atomicAdd(&cursor[bin], 1);
  ptlist[pos] = p;
}

// ---------------- fallback path (workspace too small / unexpected C) --------
__global__ __launch_bounds__(256) void lss_zero_f32(float* __restrict__ p, long long n) {
  long long i = (long long)blockIdx.x * 256 + threadIdx.x;
  if (i < n) p[i] = 0.0f;
}

__global__ __launch_bounds__(256) void lss_atomic_scatter(const float* __restrict__ x,
                                                          const float* __restrict__ geom,
                                                          float* __restrict__ out,
                                                          int np, int Cc, int perB) {
  long long idx = (long long)blockIdx.x * 256 + threadIdx.x;
  long long tot = (long long)np * Cc;
  if (idx >= tot) return;
  int p = (int)(idx / Cc);
  int c = (int)(idx - (long long)p * Cc);
  float gx = geom[3 * p + 0];
  float gy = geom[3 * p + 1];
  float gz = geom[3 * p + 2];
  int b = p / perB;
  int bin = lss_bin_of(gx, gy, gz, b);
  if (bin < 0) return;
  int local = bin - b * 40000;  // ix*200 + iy
  atomicAdd(&out[(long long)(b * Cc + c) * 40000 + local], x[idx]);
}

// ---------------------------------------------------------------------------
extern "C" void kernel_launch(void* const* d_in, const int* in_sizes, int n_in,
                              void* d_out, int out_size, void* d_ws, size_t ws_size,
                              hipStream_t stream) {
  (void)n_in;
  const float* x    = (const float*)d_in[0];
  const float* geom = (const float*)d_in[1];
  float* out = (float*)d_out;

  const long long np_ll = (long long)in_sizes[1] / 3;  // points
  const int np   = (int)np_ll;
  const int Cc   = (int)((long long)in_sizes[0] / np_ll);                    // 64
  const int Bv   = (int)((long long)out_size / ((long long)Cc * 200 * 200)); // 8
  const int perB = np / Bv;
  const int nbins = Bv * 40000;                         // 320000
  const int nblk1 = (nbins + 1023) / 1024;              // 313

  // carve workspace (~16.4 MB total; resident in the 192 MB L2)
  size_t woff = 0;
  auto carve = [&](size_t bytes) -> char* {
    char* p = (char*)d_ws + woff;
    woff = (woff + bytes + 255) & ~(size_t)255;
    return p;
  };
  int* pbin   = (int*)carve((size_t)np * 4);
  int* ptlist = (int*)carve((size_t)np * 4);
  int* counts = (int*)carve((size_t)nbins * 4);
  int* lexcl  = (int*)carve((size_t)nbins * 4);
  int* offs   = (int*)carve((size_t)nbins * 4);
  int* cursor = (int*)carve((size_t)nbins * 4);
  int* bsum   = (int*)carve((size_t)nblk1 * 4);
  int* boff   = (int*)carve((size_t)nblk1 * 4);

  const bool main_ok = (Cc == 64) && (nblk1 <= 512) && (woff <= ws_size);

  if (main_ok) {
    lss_zero_i32 <<<(nbins + 255) / 256, 256, 0, stream>>>(counts, nbins);
    lss_bin_count<<<(np + 255) / 256,    256, 0, stream>>>(geom, pbin, counts, np, perB);
    lss_scan1    <<<nblk1,               256, 0, stream>>>(counts, lexcl, bsum, nbins);
    lss_scan2    <<<1,                   512, 0, stream>>>(bsum, boff, nblk1);
    lss_scan3    <<<(nbins + 255) / 256, 256, 0, stream>>>(lexcl, boff, offs, cursor, nbins);
    lss_scatter  <<<(np + 255) / 256,    256, 0, stream>>>(pbin, cursor, ptlist, np);
    lss_gather   <<<Bv * 200,            256, 0, stream>>>(x, ptlist, offs, counts, out);
  } else {
    long long tot = (long long)out_size;
    lss_zero_f32<<<(int)((tot + 255) / 256), 256, 0, stream>>>(out, tot);
    long long work = (long long)np * Cc;
    lss_atomic_scatter<<<(int)((work + 255) / 256), 256, 0, stream>>>(x, geom, out, np, Cc, perB);
  }
}